// ColProcessor_5634997092787
// MI455X (gfx1250) — compile-verified
//
#include <hip/hip_runtime.h>
#include <hip/hip_bf16.h>
#include <float.h>

// ---------------------------------------------------------------------------
// Problem constants (from the reference)
// ---------------------------------------------------------------------------
#define COLSEL   3
#define KSEL     5
#define NAN_FILL 1.0e10f
#define BIGV     1.0e30f
#define NFEAT    32
#define TPB      256
#define TILE     (TPB * 4)   // floats staged per pipeline step (256 thr x float4)

typedef unsigned long long u64;

// ---------------------------------------------------------------------------
// gfx1250 async global->LDS DMA (ASYNCcnt) with guarded fallback
// ---------------------------------------------------------------------------
#if defined(__has_builtin)
#  if __has_builtin(__builtin_amdgcn_global_load_async_to_lds_b128) && \
      __has_builtin(__builtin_amdgcn_global_load_async_to_lds_b32)
#    define HAVE_ASYNC_LDS 1
#  endif
#  if __has_builtin(__builtin_amdgcn_s_wait_asynccnt)
#    define HAVE_WAIT_ASYNC 1
#  endif
#endif
#ifndef HAVE_ASYNC_LDS
#define HAVE_ASYNC_LDS 0
#endif
#ifndef HAVE_WAIT_ASYNC
#define HAVE_WAIT_ASYNC 0
#endif

#if HAVE_ASYNC_LDS
// builtin param types (confirmed via clang diagnostics):
//   b128: (int __vector(4) addrspace(1)*, int __vector(4) addrspace(3)*, Ii, Ii)
//   b32 : (int             addrspace(1)*, int             addrspace(3)*, Ii, Ii)
typedef int i32x4 __attribute__((vector_size(16)));
typedef __attribute__((address_space(1))) i32x4* g_v4i_p;
typedef __attribute__((address_space(3))) i32x4* l_v4i_p;
typedef __attribute__((address_space(1))) int*   g_i32_p;
typedef __attribute__((address_space(3))) int*   l_i32_p;
#endif

__device__ __forceinline__ void stage_tile(const float* __restrict__ gdist,
                                           const unsigned char* __restrict__ gdok,
                                           float* ldist, unsigned char* ldok) {
#if HAVE_ASYNC_LDS
  // per-lane: copies 16 bytes from global gdist to LDS ldist, 4 bytes from global gdok to LDS ldok
  __builtin_amdgcn_global_load_async_to_lds_b128((g_v4i_p)gdist, (l_v4i_p)ldist, 0, 0);
  __builtin_amdgcn_global_load_async_to_lds_b32 ((g_i32_p)gdok,  (l_i32_p)ldok,  0, 0);
#else
  *(float4*)ldist = *(const float4*)gdist;
  *(uchar4*)ldok  = *(const uchar4*)gdok;
#endif
}

__device__ __forceinline__ void wait_async_le(int keep2) {
  // keep2 != 0 -> allow the 2 just-issued (next-buffer) ops to stay in flight
#if HAVE_WAIT_ASYNC
  if (keep2) __builtin_amdgcn_s_wait_asynccnt(2);
  else       __builtin_amdgcn_s_wait_asynccnt(0);
#else
  if (keep2) asm volatile("s_wait_asynccnt 2" ::: "memory");
  else       asm volatile("s_wait_asynccnt 0" ::: "memory");
#endif
  asm volatile("" ::: "memory");  // compiler barrier: don't hoist LDS reads
}

// ---------------------------------------------------------------------------
// Top-K on packed u64 keys: key = (bits(dm) << 32) | index.
// dm is always a non-negative float (uniform distances; NaN->1e10, !ok->1e30),
// so IEEE bit ordering == value ordering, and ties break on smaller index —
// exactly jax.lax.top_k's tie semantics.  One v_cmp_lt_u64 per comparison.
// ---------------------------------------------------------------------------
__device__ __forceinline__ void insert5(u64 key, u64 v[KSEL]) {
  if (key >= v[KSEL - 1]) return;
  v[KSEL - 1] = key;
#pragma unroll
  for (int k = KSEL - 1; k > 0; --k) {
    if (v[k] < v[k - 1]) { u64 tmp = v[k]; v[k] = v[k - 1]; v[k - 1] = tmp; }
  }
}

__device__ __forceinline__ u64 make_key(float x, bool dok, int j) {
  const unsigned nan_hi = __float_as_uint(NAN_FILL);
  const unsigned big_hi = __float_as_uint(BIGV);
  const unsigned bx = __float_as_uint(x);
  const bool nn = __builtin_isnan(x);
  const unsigned hi = dok ? (nn ? nan_hi : bx) : big_hi;
  return ((u64)hi << 32) | (unsigned)j;
}

// ---------------------------------------------------------------------------
// Main kernel (placed first so the disasm snippet shows the async DMA path):
// one block per missing row — async-staged scan + top-5 + finalize
// ---------------------------------------------------------------------------
__global__ __launch_bounds__(TPB)
void knn_col_scan(const float* __restrict__ dist,               // (nq, nfit)
                  const unsigned char* __restrict__ mask,       // (nq, 32) bool
                  const int* __restrict__ dist_idx_map,         // (nq)
                  const int* __restrict__ row_missing_idx,      // (nrows)
                  const float* __restrict__ fitX,               // (nfit,32)
                  const unsigned char* __restrict__ mask_fit,   // (nfit,32) bool
                  const unsigned char* __restrict__ donor_ok,   // (nfit) packed
                  const float* __restrict__ col_mean_p,         // [1]
                  float* __restrict__ out,                      // (nq,32)
                  int nfit) {
  const int t = threadIdx.x;
  const int r = row_missing_idx[blockIdx.x];

  // Rows whose column-3 value isn't missing keep X's value (already copied).
  if (!mask[(size_t)r * NFEAT + COLSEL]) return;

  __shared__ __align__(16) float         s_dist[2][TILE];
  __shared__ __align__(16) unsigned char s_dok[2][TILE];
  __shared__ u64 s_k[TPB * KSEL];
  __shared__ int s_any;
  if (t == 0) s_any = 0;
  __syncthreads();

  const float* __restrict__ drow = dist + (size_t)dist_idx_map[r] * nfit;

  u64 v[KSEL];
#pragma unroll
  for (int k = 0; k < KSEL; ++k) v[k] = ~0ull;   // sentinel: worse than any real key
  int any_valid = 0;

  const int nvec  = nfit & ~(TILE - 1);   // full pipeline tiles
  const int niter = nvec / TILE;

  // ---- double-buffered async pipeline over the distance row ----
  if (niter > 0) {
    stage_tile(drow + t * 4, donor_ok + t * 4,
               &s_dist[0][t * 4], &s_dok[0][t * 4]);
    for (int it = 0; it < niter; ++it) {
      const int cur = it & 1;
      const int has_next = (it + 1 < niter);
      if (has_next) {
        const int jn = (it + 1) * TILE + t * 4;
        stage_tile(drow + jn, donor_ok + jn,
                   &s_dist[cur ^ 1][t * 4], &s_dok[cur ^ 1][t * 4]);
      }
      wait_async_le(has_next);

      const float4 dv = *(const float4*)&s_dist[cur][t * 4];
      const uchar4 ov = *(const uchar4*)&s_dok[cur][t * 4];
      const int jb = it * TILE + t * 4;
      const float xs[4] = {dv.x, dv.y, dv.z, dv.w};
      const unsigned char os[4] = {ov.x, ov.y, ov.z, ov.w};
#pragma unroll
      for (int c = 0; c < 4; ++c) {
        const bool dok = os[c] != 0;
        if (dok && !__builtin_isnan(xs[c])) any_valid = 1;
        insert5(make_key(xs[c], dok, jb + c), v);
      }
    }
  }
  // ---- scalar tail (none for nfit=16384, kept for generality) ----
  for (int j = nvec + t; j < nfit; j += TPB) {
    const float x  = drow[j];
    const bool dok = donor_ok[j] != 0;
    if (dok && !__builtin_isnan(x)) any_valid = 1;
    insert5(make_key(x, dok, j), v);
  }

  if (any_valid) atomicOr(&s_any, 1);

  // ---- tree-merge 256 sorted 5-lists in LDS (8 levels) ----
#pragma unroll
  for (int k = 0; k < KSEL; ++k) s_k[t * KSEL + k] = v[k];
  __syncthreads();
  for (int s = TPB / 2; s > 0; s >>= 1) {
    if (t < s) {
      const u64* ok = &s_k[(t + s) * KSEL];
      for (int k = 0; k < KSEL; ++k) {           // partner list is sorted:
        const u64 o = ok[k];
        if (o >= v[KSEL - 1]) break;
        insert5(o, v);
      }
#pragma unroll
      for (int k = 0; k < KSEL; ++k) s_k[t * KSEL + k] = v[k];
    }
    __syncthreads();
  }

  // ---- finalize: gather 5 donors, masked weighted mean ----
  if (t == 0) {
    float acc = 0.0f, wsum = 0.0f;
#pragma unroll
    for (int k = 0; k < KSEL; ++k) {
      const int id = (int)(v[k] & 0xffffffffu);
      const float fv = fitX[(size_t)id * NFEAT + COLSEL];
      const float w  = mask_fit[(size_t)id * NFEAT + COLSEL] ? 0.0f : 1.0f;
      acc  += fv * w;
      wsum += w;
    }
    const float knn = acc / (wsum == 0.0f ? 1.0f : wsum);
    const float val = s_any ? knn : col_mean_p[0];   // all_nan -> column mean
    out[(size_t)r * NFEAT + COLSEL] = val;
  }
}

// ---------------------------------------------------------------------------
// Kernel 1: pack donor_ok column + compute fallback column mean (one block)
// ---------------------------------------------------------------------------
__global__ __launch_bounds__(TPB)
void stats_kernel(const unsigned char* __restrict__ non_missing, // (nfit,32) bool
                  const unsigned char* __restrict__ mask_fit,    // (nfit,32) bool
                  const float* __restrict__ fitX,                // (nfit,32) f32
                  float* __restrict__ ws_mean,                   // [1]
                  unsigned char* __restrict__ ws_donor,          // [nfit]
                  int nfit) {
  __shared__ float s_sum[TPB];
  __shared__ int   s_cnt[TPB];
  const int t = threadIdx.x;
  float sum = 0.0f; int cnt = 0;
  for (int j = t; j < nfit; j += TPB) {
    ws_donor[j] = non_missing[(size_t)j * NFEAT + COLSEL];
    if (!mask_fit[(size_t)j * NFEAT + COLSEL]) {
      sum += fitX[(size_t)j * NFEAT + COLSEL];
      cnt += 1;
    }
  }
  s_sum[t] = sum; s_cnt[t] = cnt;
  __syncthreads();
  for (int s = TPB / 2; s > 0; s >>= 1) {
    if (t < s) { s_sum[t] += s_sum[t + s]; s_cnt[t] += s_cnt[t + s]; }
    __syncthreads();
  }
  if (t == 0) {
    float ms = (float)s_cnt[0];
    ws_mean[0] = s_sum[0] / (ms > 0.0f ? ms : 1.0f);
  }
}

// ---------------------------------------------------------------------------
// Kernel 2: out = X (vectorized copy)
// ---------------------------------------------------------------------------
__global__ __launch_bounds__(TPB)
void copy_kernel(const float4* __restrict__ src, float4* __restrict__ dst, int n4) {
  int i = blockIdx.x * blockDim.x + threadIdx.x;
  if (i < n4) dst[i] = src[i];
}

// ---------------------------------------------------------------------------
// Launcher
// ---------------------------------------------------------------------------
extern "C" void kernel_launch(void* const* d_in, const int* in_sizes, int n_in,
                              void* d_out, int out_size, void* d_ws, size_t ws_size,
                              hipStream_t stream) {
  const float*         X             = (const float*)d_in[0];
  const float*         dist_chunk    = (const float*)d_in[1];
  const unsigned char* non_missing   = (const unsigned char*)d_in[2]; // bool
  const unsigned char* mask_fit      = (const unsigned char*)d_in[3]; // bool
  const int*           dist_idx_map  = (const int*)d_in[4];
  const unsigned char* mask          = (const unsigned char*)d_in[5]; // bool
  const int*           row_missing   = (const int*)d_in[6];
  const float*         fitX          = (const float*)d_in[7];
  float*               out           = (float*)d_out;

  const int nq    = in_sizes[0] / NFEAT;
  const int nfit  = in_sizes[7] / NFEAT;
  const int nrows = in_sizes[6];

  // workspace layout: [0..63] col_mean (float @0), [64 .. 64+nfit) donor_ok bytes
  float*         ws_mean  = (float*)d_ws;
  unsigned char* ws_donor = (unsigned char*)d_ws + 64;

  stats_kernel<<<1, TPB, 0, stream>>>(non_missing, mask_fit, fitX, ws_mean, ws_donor, nfit);

  const int n4 = (nq * NFEAT) / 4;
  copy_kernel<<<(n4 + TPB - 1) / TPB, TPB, 0, stream>>>((const float4*)X, (float4*)out, n4);

  knn_col_scan<<<nrows, TPB, 0, stream>>>(dist_chunk, mask, dist_idx_map, row_missing,
                                          fitX, mask_fit, ws_donor, ws_mean, out, nfit);
}